// MyConv_77180562309490
// MI455X (gfx1250) — compile-verified
//
#include <hip/hip_runtime.h>
#include <stdint.h>

typedef __attribute__((ext_vector_type(2))) float v2f;
typedef __attribute__((ext_vector_type(8))) float v8f;

#define D 64
#define NTYPES 2

// Non-returning IEEE f32 max atomic: pins GLOBAL_ATOMIC_MAX_NUM_F32 (CDNA5).
// Tracked on STOREcnt; implicit wait at s_endpgm.
__device__ __forceinline__ void atomic_fmax_f32(float* addr, float val) {
    asm volatile("global_atomic_max_num_f32 %0, %1, off"
                 :
                 : "v"((unsigned long long)(uintptr_t)addr), "v"(val)
                 : "memory");
}

// ---------------- Phase 0: init out to -inf ----------------
__global__ void init_out_kernel(float* __restrict__ out, int n) {
    int i = blockIdx.x * blockDim.x + threadIdx.x;
    if (i < n) out[i] = -__builtin_inff();
}

// ---------------- Phase 1: Y[t] = X @ W[t] + b[t] via f32 WMMA ----------------
// One wave computes a 16-row strip for one weight type, all 64 output cols.
// K=64 processed as 16 WMMA steps of K=4 (V_WMMA_F32_16X16X4_F32).
__global__ void __launch_bounds__(256)
hetero_linear_wmma(const float* __restrict__ x,
                   const float* __restrict__ W,   // [2,64,64]
                   const float* __restrict__ b,   // [2,64]
                   float* __restrict__ Y,         // [2,N,64] in workspace
                   int nNodes, int nRowBlocks) {
    // Stage both weight matrices (32 KB) in LDS once per workgroup.
    __shared__ float sW[NTYPES * D * D];
    {
        const float4* srcv = (const float4*)W;
        float4* dstv = (float4*)sW;
        for (int i = threadIdx.x; i < NTYPES * D * D / 4; i += blockDim.x)
            dstv[i] = srcv[i];
    }
    __syncthreads();

    const int lane = threadIdx.x & 31;
    const int gw   = blockIdx.x * (blockDim.x >> 5) + (threadIdx.x >> 5);
    const int type = gw & 1;
    const int rb   = gw >> 1;
    if (rb >= nRowBlocks) return;   // uniform per wave: EXEC stays all-1s for WMMA

    const int rowInTile = lane & 15;
    const int col       = lane & 15;
    const int kbase     = (lane >> 4) << 1;   // lanes 0-15: K+{0,1}; lanes 16-31: K+{2,3}

    long grow = (long)rb * 16 + rowInTile;
    if (grow >= nNodes) grow = nNodes - 1;    // clamp loads in tail block
    const float* __restrict__ xrow = x + grow * D;
    const float* __restrict__ wt   = sW + type * D * D;

    v8f acc0 = {}, acc1 = {}, acc2 = {}, acc3 = {};

#pragma unroll
    for (int kk = 0; kk < 16; ++kk) {
        const int k0 = kk * 4 + kbase;
        // A-matrix 16x4 f32: VGPR0 = K+0 (lanes0-15)/K+2 (lanes16-31), VGPR1 = K+1/K+3
        v2f A;
        A.x = xrow[k0];
        A.y = xrow[k0 + 1];

        const float* __restrict__ w0 = wt + (long)k0 * D;   // row K+kbase
        v2f B0, B1, B2, B3;
        B0.x = w0[col];      B0.y = w0[col + D];        // (K, K+1) same col: 2-addr pair
        B1.x = w0[col + 16]; B1.y = w0[col + 16 + D];
        B2.x = w0[col + 32]; B2.y = w0[col + 32 + D];
        B3.x = w0[col + 48]; B3.y = w0[col + 48 + D];

        acc0 = __builtin_amdgcn_wmma_f32_16x16x4_f32(false, A, false, B0, (short)0, acc0, false, false);
        acc1 = __builtin_amdgcn_wmma_f32_16x16x4_f32(false, A, false, B1, (short)0, acc1, false, false);
        acc2 = __builtin_amdgcn_wmma_f32_16x16x4_f32(false, A, false, B2, (short)0, acc2, false, false);
        acc3 = __builtin_amdgcn_wmma_f32_16x16x4_f32(false, A, false, B3, (short)0, acc3, false, false);
    }

    // C/D layout: VGPR v -> M = v + (lane>=16 ? 8 : 0), N = lane&15
    const int mhalf = (lane >> 4) * 8;
    const float bb0 = b[type * D + col];
    const float bb1 = b[type * D + col + 16];
    const float bb2 = b[type * D + col + 32];
    const float bb3 = b[type * D + col + 48];
    const long tybase = (long)type * (long)nNodes * D;

#pragma unroll
    for (int v = 0; v < 8; ++v) {
        const long row = (long)rb * 16 + mhalf + v;
        if (row < nNodes) {
            float* __restrict__ yr = Y + tybase + row * D;
            yr[col]      = acc0[v] + bb0;
            yr[col + 16] = acc1[v] + bb1;
            yr[col + 32] = acc2[v] + bb2;
            yr[col + 48] = acc3[v] + bb3;
        }
    }
}

// ---------------- Phase 2: per-edge gather + scatter-max ----------------
// Half-wave (16 lanes) per edge: each lane gathers a float4 (b128) of the
// message row and issues 4 native f32 max-atomics. One wave = 2 edges.
__global__ void __launch_bounds__(256)
edge_scatter_max(const int* __restrict__ ei,   // [2,E] (src row 0, dst row 1)
                 const int* __restrict__ ea,   // [E] edge types
                 const float* __restrict__ Y,  // [2,N,64]
                 float* __restrict__ out,      // [N,64]
                 int E, long typeStride) {
    const int lane   = threadIdx.x & 31;
    const int half   = lane >> 4;        // which edge of this wave's pair
    const int l      = lane & 15;        // lane within half-wave
    const long gwave = (long)((blockIdx.x * blockDim.x + threadIdx.x) >> 5);
    const long nwaves = (long)((gridDim.x * blockDim.x) >> 5);

    for (long e0 = gwave * 2; e0 < E; e0 += nwaves * 2) {
        const long e  = e0 + half;
        const bool valid = (e < E);
        const long ec = valid ? e : (long)(E - 1);

        const int src = ei[ec];
        const int dst = ei[(long)E + ec];
        const int t   = ea[ec];

        const float4* __restrict__ yrow =
            (const float4*)(Y + (long)t * typeStride + (long)src * D);
        const float4 v = yrow[l];   // global_load_b128

        if (valid) {
            float* __restrict__ orow = out + (long)dst * D + 4 * l;
            atomic_fmax_f32(orow + 0, v.x);
            atomic_fmax_f32(orow + 1, v.y);
            atomic_fmax_f32(orow + 2, v.z);
            atomic_fmax_f32(orow + 3, v.w);
        }
    }
}

// ---------------- Phase 3: nodes with no incoming edges -> 0 ----------------
__global__ void finalize_out(float* __restrict__ out, int n) {
    int i = blockIdx.x * blockDim.x + threadIdx.x;
    if (i < n) {
        float v = out[i];
        if (v == -__builtin_inff()) out[i] = 0.0f;
    }
}

extern "C" void kernel_launch(void* const* d_in, const int* in_sizes, int n_in,
                              void* d_out, int out_size, void* d_ws, size_t ws_size,
                              hipStream_t stream) {
    const float* x  = (const float*)d_in[0];   // [N,64]
    const float* W  = (const float*)d_in[1];   // [2,64,64]
    const float* b  = (const float*)d_in[2];   // [2,64]
    const int*   ei = (const int*)d_in[3];     // [2,E]
    const int*   ea = (const int*)d_in[4];     // [E]

    const int N = in_sizes[0] / D;
    const int E = in_sizes[3] / 2;

    float* out = (float*)d_out;
    float* Y   = (float*)d_ws;                 // [2,N,64] = 51.2 MB scratch

    const int outElems = N * D;

    // Phase 0: out = -inf
    init_out_kernel<<<(outElems + 255) / 256, 256, 0, stream>>>(out, outElems);

    // Phase 1: Y[t] = X @ W[t] + b[t]
    const int nRowBlocks = (N + 15) / 16;
    const int totalWaves = nRowBlocks * NTYPES;
    const int blocks1    = (totalWaves + 7) / 8;   // 8 waves per 256-thread block
    hetero_linear_wmma<<<blocks1, 256, 0, stream>>>(x, W, b, Y, N, nRowBlocks);

    // Phase 2: scatter-max (half-wave per edge, 2 edges per wave)
    const int blocks2 = (E + 15) / 16;
    edge_scatter_max<<<blocks2, 256, 0, stream>>>(ei, ea, Y, out, E, (long)N * D);

    // Phase 3: -inf -> 0
    finalize_out<<<(outElems + 255) / 256, 256, 0, stream>>>(out, outElems);
}